// DisplaceChannel_62088047231333
// MI455X (gfx1250) — compile-verified
//
#include <hip/hip_runtime.h>
#include <hip/hip_bf16.h>
#include <math.h>

// Problem constants (from reference)
#define Bv   16
#define Cv   384
#define Hv   64
#define Wv   64
#define Gv   48          // NUM_POS
#define CPP  (Cv / Gv)   // 8 channels per group
#define NPIX (Hv * Wv)   // 4096 pixels per channel image
#define NTOT ((size_t)Bv * Cv * NPIX)  // 25,165,824 elements per output tensor

typedef __attribute__((ext_vector_type(4))) unsigned int u32x4;
typedef __attribute__((ext_vector_type(8))) unsigned int u32x8;
typedef __attribute__((ext_vector_type(4))) float        f32x4;

// ---------------------------------------------------------------------------
// Prep kernel: per group g compute integer shift rnd(offset) and the
// normalized 3x3 Gaussian from the fractional part. 48 lanes of work.
// Layout in d_ws: int shifts[48][2] (ox,oy) then float kerns[48][9].
// ---------------------------------------------------------------------------
__global__ __launch_bounds__(64) void prep_kernel(const float* __restrict__ offset,
                                                  int* __restrict__ shifts,
                                                  float* __restrict__ kerns) {
    int g = threadIdx.x;
    if (g >= Gv) return;
    float ox = offset[2 * g + 0];
    float oy = offset[2 * g + 1];
    float rx = rintf(ox);   // round-half-even == jnp.round; frac in (-0.45,0.45) anyway
    float ry = rintf(oy);
    shifts[2 * g + 0] = (int)rx;
    shifts[2 * g + 1] = (int)ry;
    float fx = ox - rx;
    float fy = oy - ry;
    const float inv2s2 = 1.0f / (2.0f * 0.5f * 0.5f);  // SIGMA = 0.5 -> 2.0
    float k[9];
    float s = 0.0f;
    for (int iy = 0; iy < 3; ++iy) {
        for (int ix = 0; ix < 3; ++ix) {
            float dx = (float)(ix - 1) + fx;
            float dy = (float)(iy - 1) + fy;
            float v = expf(-(dx * dx + dy * dy) * inv2s2);
            k[iy * 3 + ix] = v;
            s += v;
        }
    }
    float inv = 1.0f / s;
    for (int t = 0; t < 9; ++t) kerns[g * 9 + t] = k[t] * inv;
}

// ---------------------------------------------------------------------------
// Fused displace + depthwise 3x3 Gaussian. One workgroup per (b,c) image.
// TDM pulls the 64x64 f32 source channel into LDS (one tensor_load_to_lds,
// TENSORcnt-tracked); compute phase reads LDS only, streams both outputs
// with non-temporal b128 stores.
// ---------------------------------------------------------------------------
__global__ __launch_bounds__(256) void displace_fused_kernel(
    const float* __restrict__ inp,
    const int*   __restrict__ shifts,
    const float* __restrict__ kerns,
    float* __restrict__ out,
    float* __restrict__ out_lo) {

    __shared__ float tile[NPIX];   // 16 KB: the source channel image

    const int bc = blockIdx.x;           // 0 .. B*C-1 (uniform)
    const int c  = bc % Cv;
    const int g  = c / CPP;

    // ---- TDM: load inp[b,c,:,:] (64x64 f32, row stride 64) into LDS ----
    if (threadIdx.x < 32) {  // single wave issues the DMA (EXEC-independent op)
        const float* src = inp + (size_t)bc * NPIX;
        unsigned long long ga = (unsigned long long)(uintptr_t)src;
        // Low 32 bits of a flat shared-aperture address are the LDS byte offset.
        unsigned lds_off = (unsigned)(uintptr_t)&tile[0];

        // D# group 0: count=1 (valid, user), lds_addr, 57-bit global addr, type=2
        u32x4 d0 = { 1u,
                     lds_off,
                     (unsigned)(ga & 0xFFFFFFFFull),
                     (unsigned)((ga >> 32) & 0x01FFFFFFull) | (2u << 30) };
        // D# group 1: mask=0, data_size=2 (4B), no pad/iterate;
        // tensor_dim0=64, tensor_dim1=64, tile_dim0=64, tile_dim1=64,
        // tensor_dim0_stride=64, tensor_dim1_stride=4096
        u32x8 d1 = { (2u << 16),          // data_size = 4 bytes
                     (64u << 16),         // tensor_dim0[15:0]
                     (64u << 16),         // tensor_dim1[15:0]
                     (64u << 16),         // tile_dim0
                     64u,                 // tile_dim1
                     64u,                 // tensor_dim0_stride[31:0]
                     (4096u << 16),       // tensor_dim1_stride[15:0]
                     0u };
        asm volatile("tensor_load_to_lds %0, %1" :: "s"(d0), "s"(d1) : "memory");
        __builtin_amdgcn_s_wait_tensorcnt(0);
    }
    __syncthreads();

    // ---- per-block uniform parameters (scalar loads) ----
    const int ox = shifts[2 * g + 0];
    const int oy = shifts[2 * g + 1];
    float kk[9];
#pragma unroll
    for (int t = 0; t < 9; ++t) kk[t] = kerns[g * 9 + t];

    const size_t base_out = (size_t)bc * NPIX;

    // 4096 pixels / 256 threads: each thread does 4 iterations of 4
    // consecutive pixels -> coalesced 128-bit NT stores.
#pragma unroll
    for (int i = 0; i < 4; ++i) {
        const int p  = i * 1024 + (int)threadIdx.x * 4;
        const int y  = p >> 6;
        const int x0 = p & 63;

        float vd[4], vl[4];
#pragma unroll
        for (int j = 0; j < 4; ++j) {
            const int x  = x0 + j;
            // displaced value: out[y,x] = inp[y-oy, x-ox] with zero pad
            const int sy = y - oy;
            const int sx = x - ox;
            const bool okc = (sy >= 0) & (sy < Hv) & (sx >= 0) & (sx < Wv);
            vd[j] = okc ? tile[(sy << 6) + sx] : 0.0f;

            // out_LO[y,x] = sum_{ky,kx} kk * out[y+ky-1, x+kx-1]
            float acc = 0.0f;
#pragma unroll
            for (int ky = 0; ky < 3; ++ky) {
                const int yy  = y + ky - 1;
                const int syy = yy - oy;
                const bool yok = (yy >= 0) & (yy < Hv) & (syy >= 0) & (syy < Hv);
#pragma unroll
                for (int kx = 0; kx < 3; ++kx) {
                    const int xx  = x + kx - 1;
                    const int sxx = xx - ox;
                    const bool ok = yok & (xx >= 0) & (xx < Wv) & (sxx >= 0) & (sxx < Wv);
                    const float t = ok ? tile[(syy << 6) + sxx] : 0.0f;
                    acc = fmaf(kk[ky * 3 + kx], t, acc);
                }
            }
            vl[j] = acc;
        }

        f32x4 o4 = { vd[0], vd[1], vd[2], vd[3] };
        f32x4 l4 = { vl[0], vl[1], vl[2], vl[3] };
        // Write-once streams (201 MB total): keep them out of L2 with NT hints.
        __builtin_nontemporal_store(o4, (f32x4*)(out    + base_out + p));
        __builtin_nontemporal_store(l4, (f32x4*)(out_lo + base_out + p));
    }
}

// ---------------------------------------------------------------------------
// Launch
// ---------------------------------------------------------------------------
extern "C" void kernel_launch(void* const* d_in, const int* in_sizes, int n_in,
                              void* d_out, int out_size, void* d_ws, size_t ws_size,
                              hipStream_t stream) {
    const float* inp    = (const float*)d_in[0];   // [16,384,64,64] f32
    const float* offset = (const float*)d_in[1];   // [48,2] f32

    int*   shifts = (int*)d_ws;                    // 48*2 ints
    float* kerns  = (float*)d_ws + 2 * Gv;         // 48*9 floats, after shifts

    float* out    = (float*)d_out;                 // first tensor
    float* out_lo = (float*)d_out + NTOT;          // second tensor

    prep_kernel<<<1, 64, 0, stream>>>(offset, shifts, kerns);

    dim3 grid(Bv * Cv);   // 6144 workgroups, one 64x64 channel image each
    dim3 block(256);
    displace_fused_kernel<<<grid, block, 0, stream>>>(inp, shifts, kerns, out, out_lo);
}